// DiscriminativeCorrelationFilter_35579509080372
// MI455X (gfx1250) — compile-verified
//
#include <hip/hip_runtime.h>

#define D_FEAT 768
#define NB 32
#define NPOS_S 1024      // 32x32 search positions
#define NPOS_T 256       // 16x16 target positions
#define DCH 4            // channel chunks for streaming stages
#define CHD (D_FEAT / DCH) // 192
#define ALPHA 0.999f     // 1 - LR*LAMBDA_REG
#define LRATE 0.1f
#define EPS_BN 1e-5f

typedef __attribute__((ext_vector_type(2))) float v2f;
typedef __attribute__((ext_vector_type(8))) float v8f;

// workspace layout (float offsets)
#define WS_A0    0
#define WS_A1    (WS_A0 + D_FEAT)
#define WS_U     (WS_A1 + D_FEAT)
#define WS_V     (WS_U + D_FEAT)
#define WS_CONST (WS_V + D_FEAT)                 // [0]=c0, [1]=cS
#define WS_C5    (WS_CONST + 2)                  // NB scalars
#define WS_R0P   (WS_C5 + NB)                    // DCH*NB*NPOS_T
#define WS_SP    (WS_R0P + DCH * NB * NPOS_T)    // DCH*NB*NPOS_T
#define WS_OP    (WS_SP + DCH * NB * NPOS_T)     // DCH*NB*NPOS_S

// ---------------- Stage 1: per-channel BN fold + scalar constants ----------
__global__ void k_prep(const float* __restrict__ convb, const float* __restrict__ gamma,
                       const float* __restrict__ beta, const float* __restrict__ mean,
                       const float* __restrict__ var, const float* __restrict__ f0,
                       float* ws) {
    __shared__ float red[1024];
    int e = threadIdx.x; // block of 768
    float s = gamma[e] / sqrtf(var[e] + EPS_BN);
    float t = s * (convb[e] - mean[e]) + beta[e];
    ws[WS_A0 + e] = f0[e] * s;
    ws[WS_A1 + e] = s;

    // c0 = sum f0*t
    red[e] = f0[e] * t;
    if (e < 256) red[768 + e] = 0.0f;
    __syncthreads();
    for (int off = 512; off >= 1; off >>= 1) {
        if (e < off) red[e] += red[e + off];
        __syncthreads();
    }
    if (e == 0) ws[WS_CONST + 0] = red[0];
    __syncthreads();

    // cS = sum t
    red[e] = t;
    if (e < 256) red[768 + e] = 0.0f;
    __syncthreads();
    for (int off = 512; off >= 1; off >>= 1) {
        if (e < off) red[e] += red[e + off];
        __syncthreads();
    }
    if (e == 0) ws[WS_CONST + 1] = red[0];
}

// ---------------- Stage 2: u,v = [a0;a1] @ W via V_WMMA_F32_16X16X4_F32 ----
// A tile (16x4): lanes 0-15 hold K=0,1 of row M=lane; lanes 16-31 hold K=2,3.
// B tile (4x16): lanes 0-15 hold (K=0,1, N=lane); lanes 16-31 hold (K=2,3, N=lane-16).
// C tile (16x16): VGPR j holds row M=j (lanes 0-15). Row 0 -> u, row 1 -> v.
// Branchless: every lane loads a valid A pair, rows >= 2 are zeroed by mask.
__global__ void k_uv(const float* __restrict__ W, float* ws) {
    const float* a0 = ws + WS_A0;
    const float* a1 = ws + WS_A1;
    int lane = threadIdx.x;       // 32 lanes
    int n0 = blockIdx.x * 16;     // output column tile
    int nl = lane & 15;
    int koff = (lane < 16) ? 0 : 2;
    const float* arow = (nl == 1) ? a1 : a0;   // valid pointer for every lane
    float msk = (nl < 2) ? 1.0f : 0.0f;        // zero out rows M >= 2
    v8f c = {};
    for (int k0 = 0; k0 < D_FEAT; k0 += 4) {
        v2f araw = *(const v2f*)(arow + k0 + koff); // 8B-aligned b64 load
        v2f a;
        a.x = araw.x * msk;
        a.y = araw.y * msk;
        v2f b;
        b.x = W[(size_t)(k0 + koff)     * D_FEAT + n0 + nl];
        b.y = W[(size_t)(k0 + koff + 1) * D_FEAT + n0 + nl];
        c = __builtin_amdgcn_wmma_f32_16x16x4_f32(false, a, false, b, (short)0, c,
                                                  false, false);
    }
    if (lane < 16) {
        ws[WS_U + n0 + lane] = c[0]; // row M=0
        ws[WS_V + n0 + lane] = c[1]; // row M=1
    }
}

// ---------------- Stage 3: per-position target dots (partial over D chunk) --
// 64 threads, one float4 (4 positions) per thread => wave loads 512B/instr.
__global__ void k_target(const float* __restrict__ target, float* ws) {
    __shared__ float su[CHD], sv[CHD];
    int ch = blockIdx.x, b = blockIdx.y;
    int d0 = ch * CHD;
    int tid = threadIdx.x; // 64
    for (int i = tid; i < CHD; i += 64) {
        su[i] = ws[WS_U + d0 + i];
        sv[i] = ws[WS_V + d0 + i];
    }
    __syncthreads();
    int pos = tid * 4;
    const float* base = target + ((size_t)b * D_FEAT + d0) * NPOS_T + pos;
    float4 acc0 = make_float4(0.f, 0.f, 0.f, 0.f);
    float4 acc1 = make_float4(0.f, 0.f, 0.f, 0.f);
    for (int kk = 0; kk < CHD; kk += 32) {
        __builtin_prefetch(base + (size_t)(kk + 64) * NPOS_T, 0, 0);
#pragma unroll
        for (int i = 0; i < 32; ++i) {
            float4 x = *(const float4*)(base + (size_t)(kk + i) * NPOS_T);
            float wu = su[kk + i], wv = sv[kk + i];
            acc0.x = fmaf(x.x, wu, acc0.x); acc0.y = fmaf(x.y, wu, acc0.y);
            acc0.z = fmaf(x.z, wu, acc0.z); acc0.w = fmaf(x.w, wu, acc0.w);
            acc1.x = fmaf(x.x, wv, acc1.x); acc1.y = fmaf(x.y, wv, acc1.y);
            acc1.z = fmaf(x.z, wv, acc1.z); acc1.w = fmaf(x.w, wv, acc1.w);
        }
    }
    *(float4*)(ws + WS_R0P + ((size_t)ch * NB + b) * NPOS_T + pos) = acc0;
    *(float4*)(ws + WS_SP  + ((size_t)ch * NB + b) * NPOS_T + pos) = acc1;
}

// ---------------- Stage 4: Gaussian label + 5-iteration filter recurrence ---
__global__ void k_filter(const float* __restrict__ mask, float* ws) {
    __shared__ float red[256];
    int b = blockIdx.x;
    int tid = threadIdx.x;
    float m = mask[(size_t)b * NPOS_T + tid];
    float fy = (float)(tid >> 4);
    float fx = (float)(tid & 15);

    auto bsum = [&](float val) -> float {
        red[tid] = val;
        __syncthreads();
        for (int off = 128; off >= 1; off >>= 1) {
            if (tid < off) red[tid] += red[tid + off];
            __syncthreads();
        }
        float r = red[0];
        __syncthreads();
        return r;
    };

    float sm   = bsum(m);
    float smy  = bsum(m * fy);
    float smx  = bsum(m * fx);
    float msum = fmaxf(sm, 1.0f);
    float cy = smy / msum, cx = smx / msum;
    float dx = fx - cx, dy = fy - cy;
    float label = expf(-(dx * dx + dy * dy) * 0.125f); // 1/(2*SIGMA^2), SIGMA=2

    float r0   = ws[WS_CONST + 0];
    float ssum = ws[WS_CONST + 1];
#pragma unroll
    for (int ch = 0; ch < DCH; ++ch) {
        r0   += ws[WS_R0P + ((size_t)ch * NB + b) * NPOS_T + tid];
        ssum += ws[WS_SP  + ((size_t)ch * NB + b) * NPOS_T + tid];
    }

    float A = 1.0f, c = 0.0f;
#pragma unroll
    for (int it = 0; it < 5; ++it) {
        float resp = A * r0 + c * ssum;
        float contrib = ((1.0f - resp * label) > 0.0f) ? (-label * m) : 0.0f;
        float g = bsum(contrib) * (1.0f / 256.0f);
        c = ALPHA * c - LRATE * g;
        A *= ALPHA;
    }
    if (tid == 0) ws[WS_C5 + b] = c;
}

// ---------------- Stage 5a: streaming search dot (100 MB, bandwidth bound) --
// 256 threads, one float4 (4 positions) per thread => whole 32x32 image/block.
__global__ void k_search(const float* __restrict__ search, float* ws) {
    __shared__ float wc[CHD];
    int ch = blockIdx.x, b = blockIdx.y;
    int d0 = ch * CHD;
    int tid = threadIdx.x; // 256
    float A5 = 1.0f;
#pragma unroll
    for (int i = 0; i < 5; ++i) A5 *= ALPHA;
    float c5 = ws[WS_C5 + b];
    if (tid < CHD) wc[tid] = A5 * ws[WS_U + d0 + tid] + c5 * ws[WS_V + d0 + tid];
    __syncthreads();
    int pos = tid * 4;
    const float* base = search + ((size_t)b * D_FEAT + d0) * NPOS_S + pos;
    float4 acc = make_float4(0.f, 0.f, 0.f, 0.f);
    for (int kk = 0; kk < CHD; kk += 32) {
        __builtin_prefetch(base + (size_t)(kk + 64) * NPOS_S, 0, 0);
#pragma unroll
        for (int i = 0; i < 32; ++i) {
            float4 x = *(const float4*)(base + (size_t)(kk + i) * NPOS_S);
            float w = wc[kk + i];
            acc.x = fmaf(x.x, w, acc.x); acc.y = fmaf(x.y, w, acc.y);
            acc.z = fmaf(x.z, w, acc.z); acc.w = fmaf(x.w, w, acc.w);
        }
    }
    *(float4*)(ws + WS_OP + ((size_t)ch * NB + b) * NPOS_S + pos) = acc;
}

// ---------------- Stage 5b: combine chunk partials + affine constant --------
__global__ void k_out(const float* __restrict__ ws, float* __restrict__ out) {
    int idx4 = blockIdx.x * blockDim.x + threadIdx.x; // 8192 float4 elements
    int idx = idx4 * 4;
    int b = idx >> 10;
    int pos = idx & (NPOS_S - 1);
    float A5 = 1.0f;
#pragma unroll
    for (int i = 0; i < 5; ++i) A5 *= ALPHA;
    float c5 = ws[WS_C5 + b];
    float cst = A5 * ws[WS_CONST + 0] + c5 * ws[WS_CONST + 1];
    float4 acc = make_float4(cst, cst, cst, cst);
#pragma unroll
    for (int ch = 0; ch < DCH; ++ch) {
        float4 p = *(const float4*)(ws + WS_OP + ((size_t)ch * NB + b) * NPOS_S + pos);
        acc.x += p.x; acc.y += p.y; acc.z += p.z; acc.w += p.w;
    }
    *(float4*)(out + idx) = acc;
}

extern "C" void kernel_launch(void* const* d_in, const int* in_sizes, int n_in,
                              void* d_out, int out_size, void* d_ws, size_t ws_size,
                              hipStream_t stream) {
    const float* search = (const float*)d_in[0];
    const float* target = (const float*)d_in[1];
    const float* mask   = (const float*)d_in[2];
    const float* conv_w = (const float*)d_in[3];
    const float* conv_b = (const float*)d_in[4];
    const float* gamma  = (const float*)d_in[5];
    const float* beta   = (const float*)d_in[6];
    const float* mean   = (const float*)d_in[7];
    const float* var    = (const float*)d_in[8];
    const float* f0     = (const float*)d_in[9];
    float* out = (float*)d_out;
    float* ws  = (float*)d_ws;

    k_prep<<<1, D_FEAT, 0, stream>>>(conv_b, gamma, beta, mean, var, f0, ws);
    k_uv<<<D_FEAT / 16, 32, 0, stream>>>(conv_w, ws);
    k_target<<<dim3(DCH, NB), 64, 0, stream>>>(target, ws);
    k_filter<<<NB, 256, 0, stream>>>(mask, ws);
    k_search<<<dim3(DCH, NB), 256, 0, stream>>>(search, ws);
    k_out<<<(NB * NPOS_S / 4) / 256, 256, 0, stream>>>(ws, out);
}